// ProbAttention_84911503442554
// MI455X (gfx1250) — compile-verified
//
#include <hip/hip_runtime.h>
#include <hip/hip_bf16.h>
#include <math.h>

typedef __attribute__((ext_vector_type(2))) float v2f;
typedef __attribute__((ext_vector_type(8))) float v8f;

#define B_  2
#define H_  8
#define L_  4096
#define D_  64
#define SK_ 45
#define NTOP_ 45
#define BH_ (B_*H_)
#define QK_SIZE_ (BH_*NTOP_*L_)   // 2,949,120

// ---------------------------------------------------------------------------
// Kernel 1: per-query sampled scores -> M = max - logsumexp = -log(sum exp(x-max))
// One wave32 per query. Lane holds Q[q, 2*lane : 2*lane+2].
// ---------------------------------------------------------------------------
__global__ __launch_bounds__(256) void pa_sample_scores(
    const float* __restrict__ Q, const float* __restrict__ K,
    const int* __restrict__ isamp, float* __restrict__ Mws)
{
    const int tid  = threadIdx.x;
    const int lane = tid & 31;
    const int wave = tid >> 5;
    const int q    = blockIdx.x * 8 + wave;      // 0 .. BH*L-1
    const int bh   = q >> 12;                    // q / L
    const int l    = q & (L_ - 1);               // q % L

    const float2 qv = *reinterpret_cast<const float2*>(Q + ((size_t)(bh * L_ + l)) * D_ + 2 * lane);

    float m = -INFINITY;
    float ssum = 0.0f;
    const int* srow = isamp + l * SK_;

    #pragma unroll 5
    for (int s = 0; s < SK_; ++s) {
        const int kidx = srow[s];
        const float2 kv = *reinterpret_cast<const float2*>(K + ((size_t)(bh * L_ + kidx)) * D_ + 2 * lane);
        float part = qv.x * kv.x + qv.y * kv.y;
        // cross-lane sum over 32 lanes
        #pragma unroll
        for (int off = 16; off >= 1; off >>= 1)
            part += __shfl_xor(part, off, 32);
        // online softmax accumulation (all lanes redundantly)
        const float nm = fmaxf(m, part);
        ssum = ssum * __expf(m - nm) + __expf(part - nm);
        m = nm;
    }
    if (lane == 0)
        Mws[q] = -__logf(ssum);     // max - logsumexp
}

// ---------------------------------------------------------------------------
// Kernel 2: top-45 argmax per (b,h) over 4096 values. One block per bh.
// Iterative selection; tie broken toward lower index (matches lax.top_k).
// ---------------------------------------------------------------------------
__global__ __launch_bounds__(256) void pa_topk(
    const float* __restrict__ Mws, int* __restrict__ topk, float* __restrict__ out_tail)
{
    __shared__ float vals[L_];
    __shared__ float rv[256];
    __shared__ int   ri[256];

    const int bh  = blockIdx.x;
    const int tid = threadIdx.x;

    for (int i = tid; i < L_; i += 256)
        vals[i] = Mws[bh * L_ + i];
    __syncthreads();

    for (int t = 0; t < NTOP_; ++t) {
        float bv = -INFINITY;
        int   bi = 0;
        for (int i = tid; i < L_; i += 256) {
            const float v = vals[i];
            if (v > bv) { bv = v; bi = i; }   // ascending scan -> lowest idx on tie
        }
        rv[tid] = bv; ri[tid] = bi;
        __syncthreads();
        for (int str = 128; str >= 1; str >>= 1) {
            if (tid < str) {
                const float ov = rv[tid + str];
                const int   oi = ri[tid + str];
                if (ov > rv[tid] || (ov == rv[tid] && oi < ri[tid])) {
                    rv[tid] = ov; ri[tid] = oi;
                }
            }
            __syncthreads();
        }
        if (tid == 0) {
            const int win = ri[0];
            topk[bh * NTOP_ + t] = win;
            out_tail[bh * NTOP_ + t] = (float)win;  // M_top as float in d_out tail
            vals[win] = -INFINITY;
        }
        __syncthreads();
    }
}

// ---------------------------------------------------------------------------
// Kernel 3: Q_K[bh, n, k] = sum_d Q[bh, topk[n], d] * K[bh, k, d]
// WMMA f32 16x16x4 (exact f32). Each wave computes a 16(M) x 64(N) strip:
// 4 accumulator tiles, K-loop of 16 steps of 4.
//
// ISA layouts (wave32):
//   A 16x4 f32 : VGPR0 = {lanes0-15: K=0, lanes16-31: K=2}; VGPR1 = {K=1, K=3}
//     -> lane loads float2 at (k0 + 2*half) of its Q row
//   B 4x16 f32 : VGPR0 = {lanes0-15: row0, lanes16-31: row2}; VGPR1 = {row1, row3}
//     -> lane loads float2 at (k0 + 2*half) of its K row : contiguous b64 load
//   C/D 16x16  : VGPR v = {lanes0-15: M=v, lanes16-31: M=v+8}, N=lane&15
// ---------------------------------------------------------------------------
__global__ __launch_bounds__(256) void pa_gemm_wmma(
    const float* __restrict__ Q, const float* __restrict__ K,
    const int* __restrict__ topk, float* __restrict__ out)
{
    const int bh    = blockIdx.x;            // 0..15
    const int mtile = blockIdx.y;            // 0..2  (rows mtile*16 .. +15)
    const int wave  = threadIdx.x >> 5;      // 0..7
    const int lane  = threadIdx.x & 31;
    const int ngrp  = blockIdx.z * 8 + wave; // 0..63 -> cols ngrp*64 .. +63
    const int ncol0 = ngrp * 64;

    const int half  = lane >> 4;             // 0: lanes 0-15, 1: lanes 16-31
    const int l15   = lane & 15;

    // gather index for this lane's A row (clamped; rows >=45 are never stored)
    const int arow  = mtile * 16 + l15;
    const int qidx  = topk[bh * NTOP_ + (arow < NTOP_ ? arow : NTOP_ - 1)];
    const float* qrow  = Q + ((size_t)(bh * L_ + qidx)) * D_ + 2 * half;
    const float* kbase = K + ((size_t)bh * L_) * D_ + 2 * half;

    v8f acc[4];
    #pragma unroll
    for (int t = 0; t < 4; ++t) acc[t] = (v8f){0,0,0,0,0,0,0,0};

    #pragma unroll 4
    for (int k0 = 0; k0 < D_; k0 += 4) {
        // A fragment: contiguous float2 (dims k0+2*half, k0+2*half+1)
        const float2 af = *reinterpret_cast<const float2*>(qrow + k0);
        v2f a; a.x = af.x; a.y = af.y;

        #pragma unroll
        for (int t = 0; t < 4; ++t) {
            const int kcol = ncol0 + t * 16 + l15;        // key row this lane supplies
            // B fragment: rows {0,1} (half=0) or {2,3} (half=1) -> contiguous float2
            const float2 bf = *reinterpret_cast<const float2*>(kbase + (size_t)kcol * D_ + k0);
            v2f b; b.x = bf.x; b.y = bf.y;
            acc[t] = __builtin_amdgcn_wmma_f32_16x16x4_f32(
                false, a, false, b, (short)0, acc[t], false, false);
        }
    }

    // store: row = mtile*16 + v + 8*half, col = ncol0 + t*16 + l15
    float* obase = out + ((size_t)bh * NTOP_ + mtile * 16 + 8 * half) * L_ + ncol0 + l15;
    if (mtile < 2) {
        // rows 0..31: always in range -> fully unconditional, non-temporal
        #pragma unroll
        for (int t = 0; t < 4; ++t) {
            #pragma unroll
            for (int v = 0; v < 8; ++v)
                __builtin_nontemporal_store(acc[t][v], obase + (size_t)v * L_ + t * 16);
        }
    } else {
        // rows 32..47: v<5 always valid (rows <=44); v>=5 only for half==0 (rows 37..39)
        #pragma unroll
        for (int t = 0; t < 4; ++t) {
            #pragma unroll
            for (int v = 0; v < 5; ++v)
                __builtin_nontemporal_store(acc[t][v], obase + (size_t)v * L_ + t * 16);
        }
        if (half == 0) {
            #pragma unroll
            for (int t = 0; t < 4; ++t) {
                #pragma unroll
                for (int v = 5; v < 8; ++v)
                    __builtin_nontemporal_store(acc[t][v], obase + (size_t)v * L_ + t * 16);
            }
        }
    }
}

// ---------------------------------------------------------------------------
extern "C" void kernel_launch(void* const* d_in, const int* in_sizes, int n_in,
                              void* d_out, int out_size, void* d_ws, size_t ws_size,
                              hipStream_t stream) {
    const float* Q     = (const float*)d_in[0];
    const float* K     = (const float*)d_in[1];
    const int*   isamp = (const int*)d_in[2];
    // d_in[3] = n_top (45), hardcoded

    float* out = (float*)d_out;

    float* Mws  = (float*)d_ws;                               // BH*L floats = 256 KB
    int*   topk = (int*)((char*)d_ws + (size_t)BH_ * L_ * 4); // BH*45 ints

    // 1) sampled scores -> M   (one wave per query, 8 waves/block)
    pa_sample_scores<<<dim3((BH_ * L_) / 8), dim3(256), 0, stream>>>(Q, K, isamp, Mws);

    // 2) top-45 per (b,h); also writes M_top (as float) into d_out tail
    pa_topk<<<dim3(BH_), dim3(256), 0, stream>>>(Mws, topk, out + QK_SIZE_);

    // 3) WMMA f32 GEMM: (45x64) x (64x4096) per bh
    pa_gemm_wmma<<<dim3(BH_, 3, 8), dim3(256), 0, stream>>>(Q, K, topk, out);
}